// Pol2Vec_4870492914037
// MI455X (gfx1250) — compile-verified
//
#include <hip/hip_runtime.h>
#include <math.h>

// Problem constants (match reference: N, B, D, K = 2000, 2000, 32, 3)
#define N_POL  2000
#define N_BILL 2000
#define DIM    32

typedef __attribute__((ext_vector_type(2))) float v2f;
typedef __attribute__((ext_vector_type(8))) float v8f;

static constexpr int TILE            = 16;
static constexpr int NT_B            = N_BILL / TILE;      // 125
static constexpr int NT_N            = N_POL  / TILE;      // 125
static constexpr int N_TILES         = NT_B * NT_N;        // 15625
static constexpr int WAVES_PER_BLOCK = 8;                  // 256 threads, wave32
static constexpr int MAIN_BLOCKS     = (N_TILES + WAVES_PER_BLOCK - 1) / WAVES_PER_BLOCK;

// ---------------------------------------------------------------------------
// Kernel 1: per-politician Gram scalars + per-bill coefficients/norms
//   gram    = [6][N_POL]  : aa, bb, cc, ab, ac, bc
//   billpre = [3][N_BILL] : c1 = t, c2 = t^2/2, wsq = ||z_b||^2
// ---------------------------------------------------------------------------
__global__ void pol2vec_pre(const float* __restrict__ events_time,
                            const float* __restrict__ z_b,
                            const float* __restrict__ z_p,
                            float* __restrict__ gram,
                            float* __restrict__ billpre)
{
    int i = blockIdx.x * blockDim.x + threadIdx.x;
    if (i < N_POL) {
        const float* a = z_p + (size_t)i * DIM;                // z_p[0][i]
        const float* b = z_p + (size_t)(N_POL + i) * DIM;      // z_p[1][i]
        const float* c = z_p + (size_t)(2 * N_POL + i) * DIM;  // z_p[2][i]
        float aa = 0.f, bb = 0.f, cc = 0.f, ab = 0.f, ac = 0.f, bc = 0.f;
#pragma unroll 8
        for (int d = 0; d < DIM; d++) {
            float av = a[d], bv = b[d], cv = c[d];
            aa = fmaf(av, av, aa); bb = fmaf(bv, bv, bb); cc = fmaf(cv, cv, cc);
            ab = fmaf(av, bv, ab); ac = fmaf(av, cv, ac); bc = fmaf(bv, cv, bc);
        }
        gram[0 * N_POL + i] = aa;
        gram[1 * N_POL + i] = bb;
        gram[2 * N_POL + i] = cc;
        gram[3 * N_POL + i] = ab;
        gram[4 * N_POL + i] = ac;
        gram[5 * N_POL + i] = bc;
    }
    if (i < N_BILL) {
        float t = events_time[i];
        const float* w = z_b + (size_t)i * DIM;
        float q = 0.f;
#pragma unroll 8
        for (int d = 0; d < DIM; d++) q = fmaf(w[d], w[d], q);
        billpre[0 * N_BILL + i] = t;
        billpre[1 * N_BILL + i] = 0.5f * t * t;
        billpre[2 * N_BILL + i] = q;
    }
}

// ---------------------------------------------------------------------------
// Kernel 2: main. One wave per 16x16 (bill x politician) tile.
//   s[b,n] = P_n . u_b with K=96 via 24 chained V_WMMA_F32_16X16X4_F32.
//   A (16x4, bills):  lane m = lane%16; VGPR pair = K quad; coef scales w_b.
//   B (4x16, pols):   lane n = lane%16; rows of z_p segments.
//   D (16x16 f32):    VGPR r -> M = r + 8*(lane>=16); N = lane%16.
// ---------------------------------------------------------------------------
__global__ void __launch_bounds__(256)
pol2vec_main(const unsigned char* __restrict__ events,   // [N_POL][N_BILL] bool
             const float* __restrict__ beta,             // [N_BILL]
             const float* __restrict__ gamma,            // [N_POL]
             const float* __restrict__ z_b,              // [N_BILL][DIM]
             const float* __restrict__ z_p,              // [3][N_POL][DIM]
             const float* __restrict__ gram,             // [6][N_POL]
             const float* __restrict__ billpre,          // [3][N_BILL]
             float* __restrict__ partials)               // [MAIN_BLOCKS]
{
    __shared__ float red[WAVES_PER_BLOCK * 32];

    const int wave = threadIdx.x >> 5;
    const int lane = threadIdx.x & 31;
    const int tile = blockIdx.x * WAVES_PER_BLOCK + wave;

    float nll = 0.f;

    if (tile < N_TILES) {
        const int bt = tile / NT_N;
        const int nt = tile - bt * NT_N;
        const int b0 = bt * TILE;
        const int n0 = nt * TILE;

        const int lmod = lane & 15;
        const int khi  = lane >> 4;        // selects K pair within the quad

        // A-side (bills): this lane's bill row of U (u_b = coef(seg) * w_b)
        const int bA = b0 + lmod;
        const float c1A = billpre[0 * N_BILL + bA];
        const float c2A = billpre[1 * N_BILL + bA];
        const float* __restrict__ zb_row = z_b + (size_t)bA * DIM;

        // B-side (politicians): this lane's politician column of P
        const int nB = n0 + lmod;

        v8f acc = {};
#pragma unroll
        for (int k0 = 0; k0 < 3 * DIM; k0 += 4) {
            const int seg = k0 >> 5;                 // 0,1,2 : which z_p order
            const int kk  = (k0 & 31) + 2 * khi;     // K index within segment
            const float coef = (seg == 0) ? 1.f : ((seg == 1) ? c1A : c2A);

            v2f a;
            a.x = zb_row[kk]     * coef;
            a.y = zb_row[kk + 1] * coef;

            const float* __restrict__ zp_row = z_p + (size_t)(seg * N_POL + nB) * DIM;
            v2f bm;
            bm.x = zp_row[kk];
            bm.y = zp_row[kk + 1];

            // D = A x B + C   (f32, 16x16x4)
            acc = __builtin_amdgcn_wmma_f32_16x16x4_f32(
                false, a, false, bm, (short)0, acc, false, false);
        }

        // Epilogue: this lane owns column n = n0+lmod, rows b = b0 + r + 8*khi
        const int n = n0 + lmod;
        const float aa = gram[0 * N_POL + n];
        const float bb = gram[1 * N_POL + n];
        const float cc = gram[2 * N_POL + n];
        const float ab = gram[3 * N_POL + n];
        const float ac = gram[4 * N_POL + n];
        const float bc = gram[5 * N_POL + n];
        const float gam = gamma[n];
        const size_t evRow = (size_t)n * N_BILL;

#pragma unroll
        for (int r = 0; r < 8; r++) {
            const int b  = b0 + r + 8 * khi;
            const float c1  = billpre[0 * N_BILL + b];
            const float c2  = billpre[1 * N_BILL + b];
            const float wsq = billpre[2 * N_BILL + b];

            // ||pos||^2 from Gram scalars
            float quad = aa;
            quad = fmaf(c1 * c1, bb, quad);
            quad = fmaf(c2 * c2, cc, quad);
            quad = fmaf(2.f * c1, ab, quad);
            quad = fmaf(2.f * c2, ac, quad);
            quad = fmaf(2.f * c1 * c2, bc, quad);

            const float s  = acc[r];
            const float d2 = fmaxf(quad - 2.f * s + wsq, 0.f);
            const float dist = sqrtf(d2);

            const float logit = gam + beta[b] - dist;
            const float x = events[evRow + b] ? logit : -logit;
            // stable log_sigmoid(x) = min(x,0) - log1p(exp(-|x|))
            const float ls = fminf(x, 0.f) - log1pf(__expf(-fabsf(x)));
            nll -= ls;
        }
    }

    // Deterministic block reduction (fixed tree order)
    red[threadIdx.x] = nll;
    __syncthreads();
    for (int off = 128; off > 0; off >>= 1) {
        if ((int)threadIdx.x < off) red[threadIdx.x] += red[threadIdx.x + off];
        __syncthreads();
    }
    if (threadIdx.x == 0) partials[blockIdx.x] = red[0];
}

// ---------------------------------------------------------------------------
// Kernel 3: deterministic final reduction of block partials -> d_out[0]
// ---------------------------------------------------------------------------
__global__ void pol2vec_reduce(const float* __restrict__ partials,
                               int n, float* __restrict__ out)
{
    __shared__ float red[256];
    float s = 0.f;
    for (int i = threadIdx.x; i < n; i += 256) s += partials[i];
    red[threadIdx.x] = s;
    __syncthreads();
    for (int off = 128; off > 0; off >>= 1) {
        if ((int)threadIdx.x < off) red[threadIdx.x] += red[threadIdx.x + off];
        __syncthreads();
    }
    if (threadIdx.x == 0) out[0] = red[0];
}

// ---------------------------------------------------------------------------
// Launcher
// Inputs (setup_inputs order):
//   0: events [N,B] bool, 1: events_time [B] f32, 2: beta [B] f32,
//   3: gamma [N] f32, 4: z_b [B,D] f32, 5: z_p [K,N,D] f32
// Output: scalar f32 (negative log-likelihood)
// ---------------------------------------------------------------------------
extern "C" void kernel_launch(void* const* d_in, const int* in_sizes, int n_in,
                              void* d_out, int out_size, void* d_ws, size_t ws_size,
                              hipStream_t stream)
{
    (void)in_sizes; (void)n_in; (void)out_size; (void)ws_size;

    const unsigned char* events      = (const unsigned char*)d_in[0]; // jnp bool = 1 byte
    const float*         events_time = (const float*)d_in[1];
    const float*         beta        = (const float*)d_in[2];
    const float*         gamma       = (const float*)d_in[3];
    const float*         z_b         = (const float*)d_in[4];
    const float*         z_p         = (const float*)d_in[5];

    float* ws       = (float*)d_ws;
    float* gram     = ws;                        // 6*N_POL
    float* billpre  = gram + 6 * N_POL;          // 3*N_BILL
    float* partials = billpre + 3 * N_BILL;      // MAIN_BLOCKS

    const int preBlocks = (((N_POL > N_BILL) ? N_POL : N_BILL) + 255) / 256;
    pol2vec_pre<<<preBlocks, 256, 0, stream>>>(events_time, z_b, z_p, gram, billpre);

    pol2vec_main<<<MAIN_BLOCKS, WAVES_PER_BLOCK * 32, 0, stream>>>(
        events, beta, gamma, z_b, z_p, gram, billpre, partials);

    pol2vec_reduce<<<1, 256, 0, stream>>>(partials, MAIN_BLOCKS, (float*)d_out);
}